// Hpool_64252710748236
// MI455X (gfx1250) — compile-verified
//
#include <hip/hip_runtime.h>
#include <hip/hip_bf16.h>

typedef __attribute__((ext_vector_type(2))) float v2f;
typedef __attribute__((ext_vector_type(8))) float v8f;

constexpr int G_    = 64;
constexpr int N_    = 2048;
constexpr int C_    = 128;
constexpr int E_    = 1048576;
constexpr int EPG_  = E_ / G_;          // 16384
constexpr int NT_   = G_ * N_;          // 131072 nodes
constexpr int KP_   = 1024;             // kept per graph
constexpr int EA_   = E_ + NT_;         // 1179648 edges incl self loops
constexpr int CHUNK_ = 4096;
constexpr int NCH_   = EA_ / CHUNK_;    // 288 chunks exactly
constexpr float ALPHA_ = 0.2f;
constexpr float OMA_   = 0.8f;

// ---------------------------------------------------------------------------
// MLP GEMM: Y[M,128] = act(X[M,128] @ W[128,128] + b)   via V_WMMA_F32_16X16X4_F32
// 256 threads = 8 waves; each wave owns 16 rows x 128 cols (8 accumulator tiles).
// W staged in LDS pre-swizzled so each lane's B fragment {W[k][n],W[k+1][n]}
// is one aligned ds_load_b64 straight into the WMMA operand pair.
// A fragments are single global_load_b64 per K-step, software-pipelined.
// ---------------------------------------------------------------------------
__global__ __launch_bounds__(256) void hp_mlp_gemm(const float* __restrict__ X,
                                                   const float* __restrict__ W,
                                                   const float* __restrict__ bias,
                                                   float* __restrict__ Y,
                                                   int do_relu) {
  // sW viewed as v2f[64*128]: sW2[kp*128 + n] = { W[2kp][n], W[2kp+1][n] }
  __shared__ __align__(16) float sW[128 * 128];
  v2f* sW2 = (v2f*)sW;
  for (int i = threadIdx.x; i < 64 * 128; i += 256) {
    const int kp = i >> 7;
    const int n  = i & 127;
    v2f pr;
    pr.x = W[(2 * kp) * 128 + n];
    pr.y = W[(2 * kp + 1) * 128 + n];
    sW2[i] = pr;
  }
  __syncthreads();

  const int wave  = threadIdx.x >> 5;
  const int lane  = threadIdx.x & 31;
  const int lh    = lane & 15;
  const int hi    = (lane >= 16) ? 1 : 0;
  const int row0  = (blockIdx.x * 8 + wave) * 16;

  // A fragment for K-step ks: floats {k0+2hi, k0+1+2hi} of row (row0+lh),
  // i.e. v2f element 2*ks of this base pointer.
  const v2f* xa = (const v2f*)(X + (size_t)(row0 + lh) * 128 + 2 * hi);

  v8f acc[8];
#pragma unroll
  for (int t = 0; t < 8; ++t)
#pragma unroll
    for (int v = 0; v < 8; ++v) acc[t][v] = 0.0f;

  v2f a = xa[0];
#pragma unroll 4
  for (int ks = 0; ks < 32; ++ks) {
    v2f a_nxt = a;
    if (ks < 31) a_nxt = xa[2 * (ks + 1)];
    const v2f* brow = sW2 + (2 * ks + hi) * 128 + lh;
#pragma unroll
    for (int t = 0; t < 8; ++t) {
      v2f b = brow[t * 16];  // ds_load_b64 -> operand pair
      acc[t] = __builtin_amdgcn_wmma_f32_16x16x4_f32(
          false, a, false, b, (short)0, acc[t], false, false);
    }
    a = a_nxt;
  }

#pragma unroll
  for (int t = 0; t < 8; ++t) {
    const int n  = t * 16 + lh;
    const float bv = bias[n];
#pragma unroll
    for (int v = 0; v < 8; ++v) {
      const int r = row0 + v + (hi ? 8 : 0);
      float val = acc[t][v] + bv;
      if (do_relu) val = fmaxf(val, 0.0f);
      Y[(size_t)r * 128 + n] = val;
    }
  }
}

// ---------------------------------------------------------------------------
// Degree / norm
// ---------------------------------------------------------------------------
__global__ void hp_deg_init(float* __restrict__ deg) {
  int i = blockIdx.x * blockDim.x + threadIdx.x;
  if (i < NT_) deg[i] = 1.0f;  // self loop
}
__global__ void hp_deg_acc(const int* __restrict__ edst, float* __restrict__ deg) {
  int e = blockIdx.x * blockDim.x + threadIdx.x;
  if (e < E_) {
    int g = e / EPG_;
    atomicAdd(&deg[edst[e] + g * N_], 1.0f);
  }
}
__global__ void hp_dinv(const float* __restrict__ deg, float* __restrict__ dinv) {
  int i = blockIdx.x * blockDim.x + threadIdx.x;
  if (i < NT_) dinv[i] = rsqrtf(deg[i]);
}

// ---------------------------------------------------------------------------
// APPNP: hout = alpha*h0 + (1-alpha)*dinv[i]^2*hin[i]   (self loop folded here)
// then scatter over real edges: hout[d] += (1-alpha)*dinv[s]*dinv[d]*hin[s]
// ---------------------------------------------------------------------------
__global__ void hp_appnp_init(const float* __restrict__ h0,
                              const float* __restrict__ hin,
                              float* __restrict__ hout,
                              const float* __restrict__ dinv) {
  int idx = blockIdx.x * blockDim.x + threadIdx.x;  // one float4 per thread
  if (idx >= NT_ * 32) return;
  int node = idx >> 5;
  float di = dinv[node];
  float wl = OMA_ * di * di;
  float4 a = ((const float4*)h0)[idx];
  float4 b = ((const float4*)hin)[idx];
  float4 o;
  o.x = ALPHA_ * a.x + wl * b.x;
  o.y = ALPHA_ * a.y + wl * b.y;
  o.z = ALPHA_ * a.z + wl * b.z;
  o.w = ALPHA_ * a.w + wl * b.w;
  ((float4*)hout)[idx] = o;
}

__global__ __launch_bounds__(256) void hp_appnp_scatter(const float* __restrict__ hin,
                                                        float* __restrict__ hout,
                                                        const float* __restrict__ dinv,
                                                        const int* __restrict__ esrc,
                                                        const int* __restrict__ edst) {
  int wid  = (blockIdx.x * blockDim.x + threadIdx.x) >> 5;  // one wave = one edge
  int lane = threadIdx.x & 31;
  if (wid >= E_) return;
  int g = wid / EPG_;
  int s = esrc[wid] + g * N_;
  int d = edst[wid] + g * N_;
  float w = OMA_ * dinv[s] * dinv[d];
  float4 hv = ((const float4*)(hin + (size_t)s * 128))[lane];
  float* po = hout + (size_t)d * 128 + lane * 4;
  atomicAdd(po + 0, w * hv.x);
  atomicAdd(po + 1, w * hv.y);
  atomicAdd(po + 2, w * hv.z);
  atomicAdd(po + 3, w * hv.w);
}

// ---------------------------------------------------------------------------
// score = tanh(h . w / ||w||)   one wave per node
// ---------------------------------------------------------------------------
__global__ __launch_bounds__(256) void hp_score(const float* __restrict__ h,
                                                const float* __restrict__ w,
                                                float* __restrict__ score) {
  int wid  = (blockIdx.x * blockDim.x + threadIdx.x) >> 5;
  int lane = threadIdx.x & 31;
  if (wid >= NT_) return;
  float4 wv = ((const float4*)w)[lane];
  float4 hv = ((const float4*)(h + (size_t)wid * 128))[lane];
  float wn = wv.x * wv.x + wv.y * wv.y + wv.z * wv.z + wv.w * wv.w;
  float dp = hv.x * wv.x + hv.y * wv.y + hv.z * wv.z + hv.w * wv.w;
  for (int off = 16; off > 0; off >>= 1) {
    dp += __shfl_down(dp, off, 32);
    wn += __shfl_down(wn, off, 32);
  }
  if (lane == 0) score[wid] = tanhf(dp * rsqrtf(wn));
}

// ---------------------------------------------------------------------------
// Per-graph top-k via LDS bitonic sort of 2048 (val desc, idx asc)
// ---------------------------------------------------------------------------
__global__ __launch_bounds__(256) void hp_topk(const float* __restrict__ score,
                                               float* __restrict__ topval,
                                               int* __restrict__ perm,
                                               int* __restrict__ kept,
                                               int* __restrict__ out_batch,
                                               int* __restrict__ out_select) {
  __shared__ float sv[2048];
  __shared__ int   si[2048];
  const int g = blockIdx.x;
  for (int i = threadIdx.x; i < 2048; i += 256) {
    sv[i] = score[g * 2048 + i];
    si[i] = i;
  }
  __syncthreads();
  for (int k = 2; k <= 2048; k <<= 1) {
    for (int j = k >> 1; j > 0; j >>= 1) {
      for (int i = threadIdx.x; i < 2048; i += 256) {
        int l = i ^ j;
        if (l > i) {
          float vi = sv[i], vl = sv[l];
          int   ii = si[i], il = si[l];
          bool l_before_i = (vl > vi) || (vl == vi && il < ii);
          bool fwd = ((i & k) == 0);
          if (fwd ? l_before_i : !l_before_i) {
            sv[i] = vl; sv[l] = vi;
            si[i] = il; si[l] = ii;
          }
        }
      }
      __syncthreads();
    }
  }
  for (int t = threadIdx.x; t < 2048; t += 256) {
    int node = g * 2048 + si[t];
    if (t < KP_) {
      perm[g * KP_ + t]      = node;
      topval[g * KP_ + t]    = sv[t];
      out_batch[g * KP_ + t] = g;
      kept[node] = 1;
    } else {
      kept[node] = 0;
    }
  }
  if (threadIdx.x == 0) out_select[g] = 1;
}

// ---------------------------------------------------------------------------
// x_cat = h[perm] * topval
// ---------------------------------------------------------------------------
__global__ void hp_xcat(const float* __restrict__ h,
                        const int* __restrict__ perm,
                        const float* __restrict__ topval,
                        float* __restrict__ out) {
  int idx = blockIdx.x * blockDim.x + threadIdx.x;  // one float4
  if (idx >= G_ * KP_ * 32) return;
  int row = idx >> 5;
  int c4  = idx & 31;
  int src = perm[row];
  float sc = topval[row];
  float4 v = ((const float4*)(h + (size_t)src * 128))[c4];
  v.x *= sc; v.y *= sc; v.z *= sc; v.w *= sc;
  ((float4*)(out + (size_t)row * 128))[c4] = v;
}

// ---------------------------------------------------------------------------
// relabel = cumsum(kept)-1, per-graph scan + g*1024 offset
// ---------------------------------------------------------------------------
__global__ __launch_bounds__(256) void hp_relabel(const int* __restrict__ kept,
                                                  int* __restrict__ relabel) {
  __shared__ int part[256];
  const int g = blockIdx.x;
  const int base = g * 2048;
  int local[8];
  int sum = 0;
#pragma unroll
  for (int q = 0; q < 8; ++q) {
    local[q] = kept[base + threadIdx.x * 8 + q];
    sum += local[q];
  }
  part[threadIdx.x] = sum;
  __syncthreads();
  for (int off = 1; off < 256; off <<= 1) {
    int v = (threadIdx.x >= off) ? part[threadIdx.x - off] : 0;
    __syncthreads();
    part[threadIdx.x] += v;
    __syncthreads();
  }
  int run = part[threadIdx.x] - sum;  // exclusive prefix for this thread
#pragma unroll
  for (int q = 0; q < 8; ++q) {
    run += local[q];
    relabel[base + threadIdx.x * 8 + q] = g * KP_ + run - 1;
  }
}

// ---------------------------------------------------------------------------
// Stable partition of EA_ edges by valid flag: valid flags + chunk counts
// ---------------------------------------------------------------------------
__global__ __launch_bounds__(256) void hp_edge_valid(const int* __restrict__ esrc,
                                                     const int* __restrict__ edst,
                                                     const int* __restrict__ kept,
                                                     int* __restrict__ valid,
                                                     int* __restrict__ partial) {
  __shared__ int red[256];
  const int base = blockIdx.x * CHUNK_;
  int cnt = 0;
  for (int q = 0; q < 16; ++q) {
    int e = base + threadIdx.x * 16 + q;
    int v = 0;
    if (e < E_) {
      int g = e / EPG_;
      int s = esrc[e] + g * N_;
      int d = edst[e] + g * N_;
      v = (kept[s] && kept[d] && (s != d)) ? 1 : 0;
    }  // self-loop tail never valid (src==dst)
    valid[e] = v;
    cnt += v;
  }
  red[threadIdx.x] = cnt;
  __syncthreads();
  for (int off = 128; off > 0; off >>= 1) {
    if (threadIdx.x < off) red[threadIdx.x] += red[threadIdx.x + off];
    __syncthreads();
  }
  if (threadIdx.x == 0) partial[blockIdx.x] = red[0];
}

__global__ __launch_bounds__(512) void hp_scan_partials(int* __restrict__ partial,
                                                        int nb,
                                                        int* __restrict__ total) {
  __shared__ int s[512];
  int v = (threadIdx.x < nb) ? partial[threadIdx.x] : 0;
  s[threadIdx.x] = v;
  __syncthreads();
  for (int off = 1; off < 512; off <<= 1) {
    int t = (threadIdx.x >= off) ? s[threadIdx.x - off] : 0;
    __syncthreads();
    s[threadIdx.x] += t;
    __syncthreads();
  }
  if (threadIdx.x < nb) partial[threadIdx.x] = s[threadIdx.x] - v;  // exclusive
  if (threadIdx.x == nb - 1) *total = s[threadIdx.x];               // grand total
}

__global__ __launch_bounds__(256) void hp_edge_scatter(const int* __restrict__ esrc,
                                                       const int* __restrict__ edst,
                                                       const int* __restrict__ relabel,
                                                       const int* __restrict__ valid,
                                                       const int* __restrict__ partial,
                                                       const int* __restrict__ total_valid,
                                                       const float* __restrict__ edge_attr,
                                                       int* __restrict__ out_ei,
                                                       float* __restrict__ out_attr,
                                                       int* __restrict__ out_mask) {
  __shared__ int sums[256];
  const int base = blockIdx.x * CHUNK_;
  int lv[16];
  int cnt = 0;
#pragma unroll
  for (int q = 0; q < 16; ++q) {
    lv[q] = valid[base + threadIdx.x * 16 + q];
    cnt += lv[q];
  }
  sums[threadIdx.x] = cnt;
  __syncthreads();
  for (int off = 1; off < 256; off <<= 1) {
    int t = (threadIdx.x >= off) ? sums[threadIdx.x - off] : 0;
    __syncthreads();
    sums[threadIdx.x] += t;
    __syncthreads();
  }
  int vb  = partial[blockIdx.x] + (sums[threadIdx.x] - cnt);  // valid rank so far
  int tot = *total_valid;
  for (int q = 0; q < 16; ++q) {
    int e = base + threadIdx.x * 16 + q;
    int p;
    if (lv[q]) { p = vb; vb++; }
    else       { p = tot + (e - vb); }
    int s, d;
    if (e < E_) {
      int g = e / EPG_;
      s = esrc[e] + g * N_;
      d = edst[e] + g * N_;
    } else {
      s = d = e - E_;
    }
    out_ei[p]       = relabel[s];
    out_ei[EA_ + p] = relabel[d];
    out_mask[p]     = lv[q];
    float4* oa = (float4*)(out_attr + (size_t)p * 16);
    if (e < E_) {
      const float4* ia = (const float4*)(edge_attr + (size_t)e * 16);
      oa[0] = ia[0]; oa[1] = ia[1]; oa[2] = ia[2]; oa[3] = ia[3];
    } else {
      float4 z = {0.f, 0.f, 0.f, 0.f};
      oa[0] = z; oa[1] = z; oa[2] = z; oa[3] = z;
    }
  }
}

// ---------------------------------------------------------------------------
extern "C" void kernel_launch(void* const* d_in, const int* in_sizes, int n_in,
                              void* d_out, int out_size, void* d_ws, size_t ws_size,
                              hipStream_t stream) {
  (void)in_sizes; (void)n_in; (void)out_size; (void)ws_size;
  const float* x         = (const float*)d_in[0];
  const float* edge_attr = (const float*)d_in[1];
  const float* W1        = (const float*)d_in[2];
  const float* b1        = (const float*)d_in[3];
  const float* W2        = (const float*)d_in[4];
  const float* b2        = (const float*)d_in[5];
  const float* w_pool    = (const float*)d_in[6];
  const int*   edge_loc  = (const int*)d_in[7];  // [2, E]
  const int*   esrc = edge_loc;
  const int*   edst = edge_loc + E_;

  // ---- workspace carve-up ----
  char* ws = (char*)d_ws;
  size_t off = 0;
  auto take = [&](size_t bytes) { char* p = ws + off; off += (bytes + 255) & ~size_t(255); return p; };
  float* h0    = (float*)take((size_t)NT_ * 128 * 4);
  float* hA    = (float*)take((size_t)NT_ * 128 * 4);
  float* hB    = (float*)take((size_t)NT_ * 128 * 4);
  float* deg   = (float*)take((size_t)NT_ * 4);
  float* dinv  = (float*)take((size_t)NT_ * 4);
  float* score = (float*)take((size_t)NT_ * 4);
  int*   perm  = (int*)take((size_t)G_ * KP_ * 4);
  float* tval  = (float*)take((size_t)G_ * KP_ * 4);
  int*   kept  = (int*)take((size_t)NT_ * 4);
  int*   relab = (int*)take((size_t)NT_ * 4);
  int*   valid = (int*)take((size_t)EA_ * 4);
  int*   part  = (int*)take(512 * 4);
  int*   totv  = (int*)take(256);

  // ---- output carve-up (uniform 4-byte slots, return order) ----
  float* out_x      = (float*)d_out;                      // 8388608
  int*   out_ei     = (int*)d_out + 8388608;              // 2359296
  float* out_attr   = (float*)d_out + 10747904;           // 18874368
  int*   out_mask   = (int*)d_out + 29622272;             // 1179648
  int*   out_batch  = (int*)d_out + 30801920;             // 65536
  int*   out_select = (int*)d_out + 30867456;             // 64

  // 1) MLP (WMMA f32 16x16x4): hA = relu(x@W1+b1); h0 = hA@W2+b2
  hp_mlp_gemm<<<NT_ / 128, 256, 0, stream>>>(x, W1, b1, hA, 1);
  hp_mlp_gemm<<<NT_ / 128, 256, 0, stream>>>(hA, W2, b2, h0, 0);

  // 2) GCN degree norm
  hp_deg_init<<<NT_ / 256, 256, 0, stream>>>(deg);
  hp_deg_acc<<<E_ / 256, 256, 0, stream>>>(edst, deg);
  hp_dinv<<<NT_ / 256, 256, 0, stream>>>(deg, dinv);

  // 3) APPNP K=8 ping-pong (self loops folded into init pass)
  const float* hin = h0;
  float* bufs[2] = {hA, hB};
  for (int it = 0; it < 8; ++it) {
    float* hout = bufs[it & 1];
    hp_appnp_init<<<NT_ * 32 / 256, 256, 0, stream>>>(h0, hin, hout, dinv);
    hp_appnp_scatter<<<E_ * 32 / 256, 256, 0, stream>>>(hin, hout, dinv, esrc, edst);
    hin = hout;
  }
  const float* hfin = hin;  // hB after 8 iters

  // 4) scores + per-graph top-k + gating
  hp_score<<<NT_ * 32 / 256, 256, 0, stream>>>(hfin, w_pool, score);
  hp_topk<<<G_, 256, 0, stream>>>(score, tval, perm, kept, out_batch, out_select);
  hp_xcat<<<G_ * KP_ * 32 / 256, 256, 0, stream>>>(hfin, perm, tval, out_x);

  // 5) relabel + stable partition of edges (valid first) + scatter outputs
  hp_relabel<<<G_, 256, 0, stream>>>(kept, relab);
  hp_edge_valid<<<NCH_, 256, 0, stream>>>(esrc, edst, kept, valid, part);
  hp_scan_partials<<<1, 512, 0, stream>>>(part, NCH_, totv);
  hp_edge_scatter<<<NCH_, 256, 0, stream>>>(esrc, edst, relab, valid, part, totv,
                                            edge_attr, out_ei, out_attr, out_mask);
}